// BasicLayerV2_80272938762871
// MI455X (gfx1250) — compile-verified
//
#include <hip/hip_runtime.h>
#include <math.h>

// ---------------------------------------------------------------------------
// CDNA5 (gfx1250) implementation of the 2-layer CSWin-style block.
// - All GEMMs and both attention matmuls: V_WMMA_F32_16X16X4_F32 (fp32 matrix
//   pipe -> numerically matches the fp32 reference).
// - GEMM operand staging: GLOBAL_LOAD_ASYNC_TO_LDS_B128 (async global->LDS
//   engine, ASYNCcnt) instead of a VGPR round-trip.
// ---------------------------------------------------------------------------

typedef float v8f __attribute__((ext_vector_type(8)));
typedef float v2f __attribute__((ext_vector_type(2)));

#if defined(__HIP_DEVICE_COMPILE__)
#if !__has_builtin(__builtin_amdgcn_wmma_f32_16x16x4_f32)
#error "missing __builtin_amdgcn_wmma_f32_16x16x4_f32 on device toolchain"
#endif
#endif

static __device__ __forceinline__ v8f wmma4(v2f a, v2f b, v8f c) {
#if defined(__HIP_DEVICE_COMPILE__)
  // 8-arg form: (neg_a, A, neg_b, B, c_mod, C, reuse_a, reuse_b)
  return __builtin_amdgcn_wmma_f32_16x16x4_f32(false, a, false, b, (short)0, c,
                                               false, false);
#else
  return c;  // host pass only needs this to parse
#endif
}

// Async copy of 16 bytes global -> LDS (per lane). lds_off is the LDS byte
// offset (addrspace(3) address of a __shared__ object), gaddr a global VA.
static __device__ __forceinline__ void async_g2l_b128(unsigned lds_off,
                                                      const void* gaddr) {
#if defined(__HIP_DEVICE_COMPILE__)
  unsigned long long ga = (unsigned long long)gaddr;
  asm volatile("global_load_async_to_lds_b128 %0, %1, off"
               :
               : "v"(lds_off), "v"(ga)
               : "memory");
#else
  (void)lds_off; (void)gaddr;
#endif
}

static __device__ __forceinline__ void wait_async0() {
#if defined(__HIP_DEVICE_COMPILE__)
  asm volatile("s_wait_asynccnt 0x0" ::: "memory");
#endif
}

#define NTOK   25088   // B * 56 * 56
#define CC     256
#define LL     3136    // 56*56
#define WIMG   56

// ===========================================================================
// LayerNorm over C=256: one wave per token row, 8 waves / block.
// ===========================================================================
__global__ __launch_bounds__(256) void ln_kernel(
    const float* __restrict__ x, const float* __restrict__ g,
    const float* __restrict__ b, float* __restrict__ y, int nrows) {
  const int lane = threadIdx.x & 31;
  const int wid  = threadIdx.x >> 5;
  const int row  = blockIdx.x * 8 + wid;
  if (row >= nrows) return;
  const float* xr = x + (size_t)row * CC;
  float vals[8];
  float s = 0.f;
#pragma unroll
  for (int i = 0; i < 8; ++i) { vals[i] = xr[lane + i * 32]; s += vals[i]; }
#pragma unroll
  for (int m = 16; m >= 1; m >>= 1) s += __shfl_xor(s, m, 32);
  const float mean = s * (1.f / 256.f);
  float vs = 0.f;
#pragma unroll
  for (int i = 0; i < 8; ++i) { float d = vals[i] - mean; vs += d * d; }
#pragma unroll
  for (int m = 16; m >= 1; m >>= 1) vs += __shfl_xor(vs, m, 32);
  const float rstd = rsqrtf(vs * (1.f / 256.f) + 1e-5f);
  float* yr = y + (size_t)row * CC;
#pragma unroll
  for (int i = 0; i < 8; ++i) {
    int c = lane + i * 32;
    yr[c] = (vals[i] - mean) * rstd * g[c] + b[c];
  }
}

// ===========================================================================
// Generic GEMM: out[M,N] = epi( A[M,K] @ W[K,N] + bias , R )
//   epi 0: none   epi 1: exact GELU   epi 2: + R (residual)
// Block = 256 thr = 8 waves; tile 64x64; wave -> 16x32 via two f32 WMMA accs.
// Operand tiles staged with GLOBAL_LOAD_ASYNC_TO_LDS_B128 (ASYNCcnt).
// M % 64 == 0, N % 64 == 0, K % 16 == 0 for all call sites.
// ===========================================================================
__global__ __launch_bounds__(256) void gemm_wmma_f32(
    const float* __restrict__ A, const float* __restrict__ W,
    const float* __restrict__ bias, const float* __restrict__ R,
    float* __restrict__ out, int M, int N, int K, int epi) {
  __shared__ float As[64][20];  // 64 x 16, padded; row stride 80 B (16B mult)
  __shared__ float Bs[16][68];  // 16 x 64, padded; row stride 272 B (16B mult)

  const int tid  = threadIdx.x;
  const int lane = tid & 31;
  const int wid  = tid >> 5;
  const int hl   = lane >> 4;   // half of wave
  const int l16  = lane & 15;
  const int mblk = blockIdx.x * 64;
  const int nblk = blockIdx.y * 64;
  const int wm   = wid >> 1;    // 0..3
  const int wn   = wid & 1;     // 0..1

  v8f c0 = {}; v8f c1 = {};

  const int arow = tid >> 2;          // 0..63
  const int acol = (tid & 3) * 4;     // 0..12
  const int brow = tid >> 4;          // 0..15
  const int bcol = (tid & 15) * 4;    // 0..60

  const unsigned lds_a = (unsigned)(unsigned long long)&As[arow][acol];
  const unsigned lds_b = (unsigned)(unsigned long long)&Bs[brow][bcol];

  for (int kb = 0; kb < K; kb += 16) {
    // async global -> LDS staging (16B per lane per tile), ASYNCcnt tracked
    async_g2l_b128(lds_a, &A[(size_t)(mblk + arow) * K + kb + acol]);
    async_g2l_b128(lds_b, &W[(size_t)(kb + brow) * N + nblk + bcol]);
    wait_async0();
    __syncthreads();
    const int cb = wn * 32 + l16;
#pragma unroll
    for (int k0 = 0; k0 < 16; k0 += 4) {
      v2f a = *(const v2f*)&As[wm * 16 + l16][k0 + hl * 2];
      v2f b0, b1;
      b0.x = Bs[k0 + hl * 2][cb];          b0.y = Bs[k0 + hl * 2 + 1][cb];
      b1.x = Bs[k0 + hl * 2][cb + 16];     b1.y = Bs[k0 + hl * 2 + 1][cb + 16];
      c0 = wmma4(a, b0, c0);
      c1 = wmma4(a, b1, c1);
    }
    __syncthreads();
  }

  const int col0 = nblk + wn * 32 + l16;
  const float bia0 = bias[col0];
  const float bia1 = bias[col0 + 16];
#pragma unroll
  for (int r = 0; r < 8; ++r) {
    const int row = mblk + wm * 16 + r + hl * 8;
    const size_t o0 = (size_t)row * N + col0;
    float v0 = c0[r] + bia0;
    float v1 = c1[r] + bia1;
    if (epi == 1) {
      v0 = 0.5f * v0 * (1.f + erff(v0 * 0.70710678118654752f));
      v1 = 0.5f * v1 * (1.f + erff(v1 * 0.70710678118654752f));
    } else if (epi == 2) {
      v0 += R[o0];
      v1 += R[o0 + 16];
    }
    out[o0]      = v0;
    out[o0 + 16] = v1;
  }
}

// ===========================================================================
// Cross-shaped window attention + LEPE, one workgroup per (window, head).
//   split 0: windows 56x7 (hs=56, ws=7), channels [head*32 .. ) of C[0:128)
//   split 1: windows 7x56 (hs=7, ws=56), channels of C[128:256)
// 392 tokens/window (padded to 400), head dim 32, online-softmax flash loop.
// Dynamic LDS: q/k/v (400x33 each) + lepe (392x32) + P staging (8x16x17).
// ===========================================================================
#define ATTN_SMEM_FLOATS (3 * 400 * 33 + 392 * 32 + 8 * 16 * 17)
#define ATTN_SMEM_BYTES  (ATTN_SMEM_FLOATS * 4)

__global__ __launch_bounds__(256) void attn_kernel(
    const float* __restrict__ qkv,  // [B, 3136, 3, 256]
    const float* __restrict__ lw,   // [128, 3, 3] depthwise weights (layer/split)
    const float* __restrict__ lb,   // [128]
    float* __restrict__ outp,       // [B, 3136, 256] (concat target)
    int split) {
  extern __shared__ float sm[];
  float* q_lds  = sm;                   // 400*33
  float* k_lds  = q_lds + 400 * 33;
  float* v_lds  = k_lds + 400 * 33;
  float* lepe   = v_lds + 400 * 33;     // 392*32
  float* p_lds  = lepe + 392 * 32;      // 8*16*17

  const int tid  = threadIdx.x;
  const int lane = tid & 31;
  const int wid  = tid >> 5;
  const int hl   = lane >> 4;
  const int l16  = lane & 15;

  const int head  = blockIdx.x & 3;
  const int bw    = blockIdx.x >> 2;  // 0..63
  const int bb    = bw >> 3;          // batch
  const int w     = bw & 7;           // window index within batch
  const int choff = split * 128 + head * 32;
  const int hs    = split ? 7 : 56;
  const int ws    = split ? 56 : 7;

  // ---- cooperative load of Q, K, V window slices into LDS (zero-padded) ----
  for (int idx = tid; idx < 400 * 32; idx += 256) {
    const int t = idx >> 5, d = idx & 31;
    float qv = 0.f, kv = 0.f, vv = 0.f;
    if (t < 392) {
      int row, col;
      if (split == 0) { row = t / 7;  col = w * 7 + (t - row * 7); }
      else            { int i = t / 56; row = w * 7 + i; col = t - i * 56; }
      const size_t base =
          ((size_t)(bb * LL + row * WIMG + col) * 3) * CC + choff + d;
      qv = qkv[base];
      kv = qkv[base + CC];
      vv = qkv[base + 2 * CC];
    }
    q_lds[t * 33 + d] = qv;
    k_lds[t * 33 + d] = kv;
    v_lds[t * 33 + d] = vv;
  }
  __syncthreads();

  // ---- LEPE: depthwise 3x3 SAME conv over the window spatial tile ----------
  for (int idx = tid; idx < 392 * 32; idx += 256) {
    const int t = idx >> 5, d = idx & 31;
    const int i = t / ws, j = t - i * ws;
    const int ch = head * 32 + d;
    float acc = lb[ch];
#pragma unroll
    for (int ky = 0; ky < 3; ++ky)
#pragma unroll
      for (int kx = 0; kx < 3; ++kx) {
        const int ii = i + ky - 1, jj = j + kx - 1;
        if (ii >= 0 && ii < hs && jj >= 0 && jj < ws)
          acc += lw[ch * 9 + ky * 3 + kx] * v_lds[(ii * ws + jj) * 33 + d];
      }
    lepe[t * 32 + d] = acc;
  }
  __syncthreads();

  // ---- flash attention: each wave owns 16-query-row blocks -----------------
  const float scale = 0.17677669529663687f;  // 32^-0.5
  for (int qb = wid; qb < 25; qb += 8) {
    const int m0 = qb * 16;
    v8f o0 = {}; v8f o1 = {};
    float mrow[8], srow[8];
#pragma unroll
    for (int r = 0; r < 8; ++r) { mrow[r] = -1e30f; srow[r] = 0.f; }

    for (int kt = 0; kt < 25; ++kt) {
      // S tile = Q_blk (16x32) @ K^T  for key cols [kt*16, kt*16+16)
      v8f s = {};
#pragma unroll
      for (int d0 = 0; d0 < 32; d0 += 4) {
        v2f a = *(const v2f*)&q_lds[(m0 + l16) * 33 + d0 + hl * 2];
        v2f b = *(const v2f*)&k_lds[(kt * 16 + l16) * 33 + d0 + hl * 2];
        s = wmma4(a, b, s);
      }
      const bool valid = (kt * 16 + l16) < 392;

      // online softmax update (per C-tile row r; row = r + hl*8)
#pragma unroll
      for (int r = 0; r < 8; ++r) {
        float sv = valid ? s[r] * scale : -1e30f;
        float tm = sv;
#pragma unroll
        for (int m = 8; m >= 1; m >>= 1) tm = fmaxf(tm, __shfl_xor(tm, m, 32));
        const float mn    = fmaxf(mrow[r], tm);
        const float alpha = __expf(mrow[r] - mn);
        const float pe    = __expf(sv - mn);
        float rs = pe;
#pragma unroll
        for (int m = 8; m >= 1; m >>= 1) rs += __shfl_xor(rs, m, 32);
        srow[r] = srow[r] * alpha + rs;
        mrow[r] = mn;
        o0[r] *= alpha;
        o1[r] *= alpha;
        p_lds[wid * 272 + (r + hl * 8) * 17 + l16] = pe;  // C-layout -> LDS
      }

      // O += P (16x16) @ V_blk (16x32)   (P read back in A-fragment layout)
#pragma unroll
      for (int k0 = 0; k0 < 16; k0 += 4) {
        v2f a;
        a.x = p_lds[wid * 272 + l16 * 17 + k0 + hl * 2];
        a.y = p_lds[wid * 272 + l16 * 17 + k0 + hl * 2 + 1];
        const int vr = kt * 16 + k0 + hl * 2;
        v2f b0, b1;
        b0.x = v_lds[vr * 33 + l16];            b0.y = v_lds[(vr + 1) * 33 + l16];
        b1.x = v_lds[vr * 33 + 16 + l16];       b1.y = v_lds[(vr + 1) * 33 + 16 + l16];
        o0 = wmma4(a, b0, o0);
        o1 = wmma4(a, b1, o1);
      }
    }

    // ---- normalize, add LEPE, scatter back to image layout -----------------
#pragma unroll
    for (int r = 0; r < 8; ++r) {
      const int t = m0 + r + hl * 8;
      if (t < 392) {
        const float inv = 1.f / srow[r];
        int row, col;
        if (split == 0) { row = t / 7;  col = w * 7 + (t - row * 7); }
        else            { int i = t / 56; row = w * 7 + i; col = t - i * 56; }
        const size_t base = (size_t)(bb * LL + row * WIMG + col) * CC + choff;
        outp[base + l16]      = o0[r] * inv + lepe[t * 32 + l16];
        outp[base + 16 + l16] = o1[r] * inv + lepe[t * 32 + 16 + l16];
      }
    }
  }
}

// ===========================================================================
// Host side: sequence the 2 layers on `stream`.
// Workspace layout (floats):
//   buf1: 25088*1024  (qkv [uses 768 cols] then fc1 hidden)   ~102.8 MB
//   buf2: 25088*256   (LN1 / LN2 output)                       ~25.7 MB
//   buf3: 25088*256   (attention output, concat of both splits)~25.7 MB
// d_out holds the running residual x (fully written by layer-0 proj epilogue
// before ever being read, so the 0xAA poison never leaks).
// ===========================================================================
extern "C" void kernel_launch(void* const* d_in, const int* in_sizes, int n_in,
                              void* d_out, int out_size, void* d_ws,
                              size_t ws_size, hipStream_t stream) {
  (void)in_sizes; (void)n_in; (void)out_size; (void)ws_size;

  const float* x_in    = (const float*)d_in[0];
  const float* ln1_g   = (const float*)d_in[1];
  const float* ln1_b   = (const float*)d_in[2];
  const float* qkv_w   = (const float*)d_in[3];
  const float* qkv_b   = (const float*)d_in[4];
  const float* lepe_w0 = (const float*)d_in[5];
  const float* lepe_b0 = (const float*)d_in[6];
  const float* lepe_w1 = (const float*)d_in[7];
  const float* lepe_b1 = (const float*)d_in[8];
  const float* proj_w  = (const float*)d_in[9];
  const float* proj_b  = (const float*)d_in[10];
  const float* ln2_g   = (const float*)d_in[11];
  const float* ln2_b   = (const float*)d_in[12];
  const float* fc1_w   = (const float*)d_in[13];
  const float* fc1_b   = (const float*)d_in[14];
  const float* fc2_w   = (const float*)d_in[15];
  const float* fc2_b   = (const float*)d_in[16];

  float* xo   = (float*)d_out;
  float* buf1 = (float*)d_ws;                          // qkv / mlp hidden
  float* buf2 = buf1 + (size_t)NTOK * 1024;            // LN outputs
  float* buf3 = buf2 + (size_t)NTOK * CC;              // attention output

  static_assert(ATTN_SMEM_BYTES < 320 * 1024, "LDS budget");
  (void)hipFuncSetAttribute((const void*)attn_kernel,
                            hipFuncAttributeMaxDynamicSharedMemorySize,
                            ATTN_SMEM_BYTES);

  const dim3 blk(256);
  for (int l = 0; l < 2; ++l) {
    const float* xsrc = (l == 0) ? x_in : (const float*)xo;

    // LN1
    ln_kernel<<<NTOK / 8, blk, 0, stream>>>(xsrc, ln1_g + l * CC,
                                            ln1_b + l * CC, buf2, NTOK);
    // QKV GEMM: [25088,256] @ [256,768]
    gemm_wmma_f32<<<dim3(NTOK / 64, 768 / 64), blk, 0, stream>>>(
        buf2, qkv_w + (size_t)l * CC * 768, qkv_b + l * 768, nullptr, buf1,
        NTOK, 768, CC, 0);
    // attention (both cross-shaped splits; together they fill all 256 ch)
    attn_kernel<<<256, blk, ATTN_SMEM_BYTES, stream>>>(
        buf1, lepe_w0 + (size_t)l * 128 * 9, lepe_b0 + l * 128, buf3, 0);
    attn_kernel<<<256, blk, ATTN_SMEM_BYTES, stream>>>(
        buf1, lepe_w1 + (size_t)l * 128 * 9, lepe_b1 + l * 128, buf3, 1);
    // proj + residual: x = xsrc + attn @ proj_w + proj_b
    gemm_wmma_f32<<<dim3(NTOK / 64, CC / 64), blk, 0, stream>>>(
        buf3, proj_w + (size_t)l * CC * CC, proj_b + l * CC, xsrc, xo, NTOK,
        CC, CC, 2);
    // LN2
    ln_kernel<<<NTOK / 8, blk, 0, stream>>>(xo, ln2_g + l * CC, ln2_b + l * CC,
                                            buf2, NTOK);
    // fc1 + exact GELU: [25088,256] @ [256,1024]
    gemm_wmma_f32<<<dim3(NTOK / 64, 1024 / 64), blk, 0, stream>>>(
        buf2, fc1_w + (size_t)l * CC * 1024, fc1_b + l * 1024, nullptr, buf1,
        NTOK, 1024, CC, 1);
    // fc2 + residual: x = x + h @ fc2_w + fc2_b
    gemm_wmma_f32<<<dim3(NTOK / 64, CC / 64), blk, 0, stream>>>(
        buf1, fc2_w + (size_t)l * 1024 * CC, fc2_b + l * CC, xo, xo, NTOK, CC,
        1024, 2);
  }
}